// TtAttention_63359357551123
// MI455X (gfx1250) — compile-verified
//
#include <hip/hip_runtime.h>
#include <hip/hip_bf16.h>

// ---------------------------------------------------------------------------
// Decode-step GQA attention, BSZ=16, DIM=4096, 32 q-heads / 8 kv-heads,
// HEAD_DIM=128, cache length 4096 (start_pos=4095, seqlen=1). fp32 throughout.
// Projections use V_WMMA_F32_16X16X4_F32 (M=16 == batch).
// ---------------------------------------------------------------------------

typedef __attribute__((ext_vector_type(2))) float v2f;
typedef __attribute__((ext_vector_type(4))) float v4f;
typedef __attribute__((ext_vector_type(8))) float v8f;

#define DIM 4096
#define KDIM 4096
#define N_HEADS 32
#define N_KV_HEADS 8
#define HEAD_DIM 128
#define QKV_STRIDE 6144   // 4096 q + 1024 k + 1024 v per batch row
#define KV_ROW (N_KV_HEADS * HEAD_DIM)  // 1024 floats between positions

// ---------------------------------------------------------------------------
// gemm16: out[m][colBase+n] = sum_k X[m][k] * W[colBase+n][k]
//   X: 16 x 4096 (row-major, stride 4096)
//   W: (ncols) x 4096 (row-major, stride 4096) -- grid.x = ncols/16
//   out: 16 x outStride
// One 16x16 tile per block; 4 waves split K; LDS reduction of partial C.
// ---------------------------------------------------------------------------
__global__ __launch_bounds__(128) void gemm16(const float* __restrict__ X,
                                              const float* __restrict__ W,
                                              float* __restrict__ out,
                                              int outStride) {
    __shared__ float red[4 * 256];

    const int lane = threadIdx.x & 31;
    const int wave = threadIdx.x >> 5;
    const int colBase = blockIdx.x * 16;

    const int mn    = lane & 15;        // A row (batch m) / B col (n)
    const int khalf = (lane >> 4) * 2;  // K sub-offset 0 or 2 per ISA layout

    const float* __restrict__ xrow = X + (size_t)mn * KDIM;
    const float* __restrict__ wrow = W + (size_t)(colBase + mn) * KDIM;

    v8f acc = {};

    const int k_begin = wave * (KDIM / 4);
    const int k_end   = k_begin + (KDIM / 4);

#pragma unroll 8
    for (int k0 = k_begin; k0 < k_end; k0 += 4) {
        const int kk = k0 + khalf;
        v2f a = *reinterpret_cast<const v2f*>(xrow + kk);
        v2f b = *reinterpret_cast<const v2f*>(wrow + kk);
        // D = A(16x4 f32) * B(4x16 f32) + C
        acc = __builtin_amdgcn_wmma_f32_16x16x4_f32(
            /*neg_a=*/false, a, /*neg_b=*/false, b,
            /*c_mod=*/(short)0, acc, /*reuse_a=*/false, /*reuse_b=*/false);
    }

    // Stash per-wave partial C tiles in LDS, then reduce + store.
#pragma unroll
    for (int g = 0; g < 8; ++g)
        red[wave * 256 + lane * 8 + g] = acc[g];
    __syncthreads();

    for (int e = threadIdx.x; e < 256; e += 128) {
        float s = red[e] + red[256 + e] + red[512 + e] + red[768 + e];
        const int L = e >> 3;            // lane
        const int g = e & 7;             // C vgpr index
        const int m = g + 8 * (L >> 4);  // ISA C-layout: M = vgpr + 8*(lane/16)
        const int n = L & 15;            // N = lane % 16
        out[(size_t)m * outStride + colBase + n] = s;
    }
}

// ---------------------------------------------------------------------------
// RoPE on q (32 heads) and k (8 heads) rows of the qkv workspace, pos 4095.
// freqs_cos/sin: 64 entries (seqlen=1).
// ---------------------------------------------------------------------------
__global__ __launch_bounds__(256) void rope_kernel(float* __restrict__ qkv,
                                                   const float* __restrict__ cosv,
                                                   const float* __restrict__ sinv) {
    const int total = 16 * (N_HEADS + N_KV_HEADS) * (HEAD_DIM / 2); // 40960
    int tid = blockIdx.x * blockDim.x + threadIdx.x;
    if (tid >= total) return;
    const int p  = tid & 63;
    const int hh = (tid >> 6) % (N_HEADS + N_KV_HEADS);
    const int b  = tid / (64 * (N_HEADS + N_KV_HEADS));
    const int col = (hh < N_HEADS) ? (hh * HEAD_DIM + 2 * p)
                                   : (DIM + (hh - N_HEADS) * HEAD_DIM + 2 * p);
    float* ptr = qkv + (size_t)b * QKV_STRIDE + col;
    const float a = ptr[0], bb = ptr[1];
    const float c = cosv[p], s = sinv[p];
    ptr[0] = a * c - bb * s;
    ptr[1] = a * s + bb * c;
}

// ---------------------------------------------------------------------------
// Attention helpers
// ---------------------------------------------------------------------------
__device__ __forceinline__ float wave_sum32(float v) {
#pragma unroll
    for (int d = 16; d > 0; d >>= 1)
        v += __shfl_xor(v, d, 32);
    return v;
}

__device__ __forceinline__ float dot4(const v4f& a, const v4f& b) {
    return a[0] * b[0] + a[1] * b[1] + a[2] * b[2] + a[3] * b[3];
}

__device__ __forceinline__ void attn_step(const v4f& q, const v4f& kk, const v4f& vv,
                                          float& m, float& l, v4f& acc) {
    const float s    = wave_sum32(dot4(q, kk));
    const float mnew = fmaxf(m, s);
    const float corr = __expf(m - mnew);
    const float p    = __expf(s - mnew);
    l   = l * corr + p;
    acc = acc * corr + p * vv;
    m   = mnew;
}

// ---------------------------------------------------------------------------
// Flash-style decode attention. One wave (32 lanes) per (batch, head).
// Lane owns head-dims 4*lane..4*lane+3 -> b128 loads are exactly coalesced
// (32 lanes x 16B = one full 512B key/value vector). 4 positions per
// iteration: independent shuffle-reduce chains pipeline, and the serial
// softmax rescale (max/corr/exp of running state) is amortized 4x.
// Position start_pos sources k/v from the fresh projections (cache unmodified).
// ---------------------------------------------------------------------------
__global__ __launch_bounds__(32) void attn_decode(const float* __restrict__ qkv,
                                                  const float* __restrict__ cache_k,
                                                  const float* __restrict__ cache_v,
                                                  const int* __restrict__ sp_ptr,
                                                  float* __restrict__ attn_out) {
    const int bh   = blockIdx.x;
    const int b    = bh >> 5;       // / 32 heads
    const int h    = bh & 31;
    const int hkv  = h >> 2;        // n_rep = 4
    const int lane = threadIdx.x;

    const int sp = *sp_ptr;         // 4095; cached positions [0, sp) + 1 fresh

    const float scale = 0.08838834764831845f; // 1/sqrt(128)

    const float* qrow = qkv + (size_t)b * QKV_STRIDE + h * HEAD_DIM + 4 * lane;
    v4f q = *reinterpret_cast<const v4f*>(qrow);
    q *= scale;

    // element (b, j, hkv, d) at ((b*4096 + j)*8 + hkv)*128 + d
    const float* kb = cache_k + ((size_t)b * 4096 * N_KV_HEADS + hkv) * HEAD_DIM + 4 * lane;
    const float* vb = cache_v + ((size_t)b * 4096 * N_KV_HEADS + hkv) * HEAD_DIM + 4 * lane;
    const float* knew = qkv + (size_t)b * QKV_STRIDE + DIM + hkv * HEAD_DIM + 4 * lane;
    const float* vnew = qkv + (size_t)b * QKV_STRIDE + DIM + 1024 + hkv * HEAD_DIM + 4 * lane;

    float m = -3.0e38f;
    float l = 0.0f;
    v4f acc = {};

    const int nfull = sp & ~3;      // 4-wide main loop over cached positions
    int j = 0;
    for (; j < nfull; j += 4) {
        const float* kp = kb + (size_t)j * KV_ROW;
        const float* vp = vb + (size_t)j * KV_ROW;
        // 8 independent b128 loads (offsets fold into the 24-bit ioffset)
        v4f k0 = *reinterpret_cast<const v4f*>(kp);
        v4f k1 = *reinterpret_cast<const v4f*>(kp + KV_ROW);
        v4f k2 = *reinterpret_cast<const v4f*>(kp + 2 * KV_ROW);
        v4f k3 = *reinterpret_cast<const v4f*>(kp + 3 * KV_ROW);
        v4f w0 = *reinterpret_cast<const v4f*>(vp);
        v4f w1 = *reinterpret_cast<const v4f*>(vp + KV_ROW);
        v4f w2 = *reinterpret_cast<const v4f*>(vp + 2 * KV_ROW);
        v4f w3 = *reinterpret_cast<const v4f*>(vp + 3 * KV_ROW);

        // 4 independent reduction chains -> pipelined shuffles
        float s0 = wave_sum32(dot4(q, k0));
        float s1 = wave_sum32(dot4(q, k1));
        float s2 = wave_sum32(dot4(q, k2));
        float s3 = wave_sum32(dot4(q, k3));

        const float mnew = fmaxf(fmaxf(m, fmaxf(s0, s1)), fmaxf(s2, s3));
        const float corr = __expf(m - mnew);
        const float p0 = __expf(s0 - mnew);
        const float p1 = __expf(s1 - mnew);
        const float p2 = __expf(s2 - mnew);
        const float p3 = __expf(s3 - mnew);

        l   = l * corr + ((p0 + p1) + (p2 + p3));
        acc = acc * corr + p0 * w0 + p1 * w1 + p2 * w2 + p3 * w3;
        m   = mnew;
    }

    // cached-position tail (sp % 4 positions)
    for (; j < sp; ++j) {
        const v4f kk = *reinterpret_cast<const v4f*>(kb + (size_t)j * KV_ROW);
        const v4f vv = *reinterpret_cast<const v4f*>(vb + (size_t)j * KV_ROW);
        attn_step(q, kk, vv, m, l, acc);
    }

    // fresh position (start_pos): k/v from this step's projections
    {
        const v4f kk = *reinterpret_cast<const v4f*>(knew);
        const v4f vv = *reinterpret_cast<const v4f*>(vnew);
        attn_step(q, kk, vv, m, l, acc);
    }

    const v4f o = acc * (1.0f / l);
    float* dst = attn_out + (size_t)b * DIM + h * HEAD_DIM + 4 * lane;
    *reinterpret_cast<v4f*>(dst) = o;
}

// ---------------------------------------------------------------------------
// Launch: qkv projections -> rope -> attention -> output projection.
// ---------------------------------------------------------------------------
extern "C" void kernel_launch(void* const* d_in, const int* in_sizes, int n_in,
                              void* d_out, int out_size, void* d_ws, size_t ws_size,
                              hipStream_t stream) {
    const float* x  = (const float*)d_in[0];
    const float* wq = (const float*)d_in[1];
    const float* wk = (const float*)d_in[2];
    const float* wv = (const float*)d_in[3];
    const float* wo = (const float*)d_in[4];
    const float* ck = (const float*)d_in[5];
    const float* cv = (const float*)d_in[6];
    const float* fc = (const float*)d_in[7];
    const float* fs = (const float*)d_in[8];
    const int*   sp = (const int*)d_in[9];

    float* qkv  = (float*)d_ws;            // 16 x 6144
    float* attn = qkv + 16 * QKV_STRIDE;   // 16 x 4096
    float* out  = (float*)d_out;           // 16 x 4096

    // xq / xk / xv projections (out = x . W^T)
    gemm16<<<DIM / 16,  128, 0, stream>>>(x, wq, qkv,        QKV_STRIDE);
    gemm16<<<1024 / 16, 128, 0, stream>>>(x, wk, qkv + DIM,  QKV_STRIDE);
    gemm16<<<1024 / 16, 128, 0, stream>>>(x, wv, qkv + 5120, QKV_STRIDE);

    // RoPE on q and k (position start_pos)
    rope_kernel<<<(40960 + 255) / 256, 256, 0, stream>>>(qkv, fc, fs);

    // decode attention (16 batches x 32 heads = 512 waves)
    attn_decode<<<16 * N_HEADS, 32, 0, stream>>>(qkv, ck, cv, sp, attn);

    // output projection into d_out
    gemm16<<<DIM / 16, 128, 0, stream>>>(attn, wo, out, DIM);
}